// generateq_60129542144134
// MI455X (gfx1250) — compile-verified
//
#include <hip/hip_runtime.h>
#include <hip/hip_bf16.h>
#include <math.h>

typedef __attribute__((ext_vector_type(2))) float v2f;
typedef __attribute__((ext_vector_type(8))) float v8f;

#define BB      16
#define CC      256
#define HWSZ    16384   // 128*128
#define INTER   128
#define CHUNKS  32
#define CHUNK_ELEMS 512 // floats per chunk (128 threads * float4)

static __device__ __forceinline__ v8f wmma_f32_16x16x4(v2f a, v2f b, v8f c) {
  // D = A(16x4 f32) * B(4x16 f32) + C(16x16 f32), full fp32 precision
  return __builtin_amdgcn_wmma_f32_16x16x4_f32(false, a, false, b, (short)0, c,
                                               false, false);
}

// ---------------------------------------------------------------------------
// Kernel 1: fused single pass over x.
//   - per-channel partial sums (for the avg-pool)  -> partials[b][chunk][c]
//   - channel dot with W3 + exact GELU (eca x3)    -> out[b][0][pos]
// grid = (CHUNKS, B), block = 128 threads (4 waves), each thread owns 4 floats
// ---------------------------------------------------------------------------
__global__ __launch_bounds__(128) void nlb_pool_eca(
    const float* __restrict__ x, const float* __restrict__ W3,
    float* __restrict__ out, float* __restrict__ partials) {
  const int chunk = blockIdx.x;
  const int b     = blockIdx.y;
  const int tid   = threadIdx.x;
  const int wave  = tid >> 5;
  const int lane  = tid & 31;

  __shared__ float lds_part[4][CC];   // per-wave partial channel sums

  const float* xb  = x + (size_t)b * CC * HWSZ;
  const int    pos = chunk * CHUNK_ELEMS + tid * 4;

  float4 acc = {0.f, 0.f, 0.f, 0.f};
#pragma unroll 4
  for (int c = 0; c < CC; ++c) {
    const float4 v = *(const float4*)(xb + (size_t)c * HWSZ + pos);
    const float  w = W3[c];                     // uniform -> scalar load
    acc.x = fmaf(v.x, w, acc.x);
    acc.y = fmaf(v.y, w, acc.y);
    acc.z = fmaf(v.z, w, acc.z);
    acc.w = fmaf(v.w, w, acc.w);
    float s = v.x + v.y + v.z + v.w;            // pooling contribution
    s += __shfl_xor(s, 16, 32);
    s += __shfl_xor(s, 8, 32);
    s += __shfl_xor(s, 4, 32);
    s += __shfl_xor(s, 2, 32);
    s += __shfl_xor(s, 1, 32);
    if (lane == 0) lds_part[wave][c] = s;
  }
  __syncthreads();
  // 128 threads x 2 channels each: deterministic 4-way combine
  for (int c = tid; c < CC; c += 128) {
    float s = lds_part[0][c] + lds_part[1][c] + lds_part[2][c] + lds_part[3][c];
    partials[((size_t)(b * CHUNKS + chunk)) * CC + c] = s;
  }
  // exact GELU: 0.5*x*(1+erf(x/sqrt(2)))  -> out channel 0
  const float is2 = 0.70710678118654752440f;
  float4 g;
  g.x = 0.5f * acc.x * (1.f + erff(acc.x * is2));
  g.y = 0.5f * acc.y * (1.f + erff(acc.y * is2));
  g.z = 0.5f * acc.z * (1.f + erff(acc.z * is2));
  g.w = 0.5f * acc.w * (1.f + erff(acc.w * is2));
  *(float4*)(out + (size_t)b * 4 * HWSZ + pos) = g;
}

// ---------------------------------------------------------------------------
// Kernel 2: whole NL-block on the pooled vector, WMMA fp32 (16x16x4).
// 1 block, 256 threads = 8 waves. M=16 (batch), per-wave 16-wide N tile.
// Fragment layout per ISA 7.12.2:
//   A 16x4 f32: lanes 0-15 row M=l, VGPR0/1 = K0/K1; lanes 16-31 = K2/K3
//   B 4x16 f32: VGPR v, lanes 0-15 = row K=v (N=lane); lanes 16-31 = K=v+2
//   C/D 16x16 : VGPR v, lanes 0-15 -> M=v, lanes 16-31 -> M=v+8, N=lane%16
// ---------------------------------------------------------------------------
__global__ __launch_bounds__(256) void nlb_tiny(
    const float* __restrict__ partials,
    const float* __restrict__ Wg, const float* __restrict__ bg,
    const float* __restrict__ Wt, const float* __restrict__ bt,
    const float* __restrict__ Wp, const float* __restrict__ bp,
    const float* __restrict__ Wf, const float* __restrict__ bf,
    const float* __restrict__ Wz, const float* __restrict__ bz,
    const float* __restrict__ gamma, const float* __restrict__ beta,
    int* __restrict__ topk) {
  __shared__ float p_lds[BB][CC];
  __shared__ float g_lds[BB][INTER];
  __shared__ float th_lds[BB][INTER];
  __shared__ float ph_lds[BB][INTER];
  __shared__ float wy_lds[BB][CC];
  __shared__ float f_lds[BB];

  const int tid  = threadIdx.x;
  const int wave = tid >> 5;
  const int lane = tid & 31;
  const int half = lane >> 4;   // 0: K pair {0,1}; 1: K pair {2,3}
  const int l    = lane & 15;   // row (A) / column-in-tile (B,D)

  // ---- pooled p = (sum of 32 chunk partials)/16384, fixed order ----
  for (int e = tid; e < BB * CC; e += 256) {
    const int b = e >> 8, c = e & 255;
    float s = 0.f;
    for (int k = 0; k < CHUNKS; ++k)
      s += partials[(size_t)(b * CHUNKS + k) * CC + c];
    p_lds[b][c] = s * (1.0f / (float)HWSZ);
  }
  __syncthreads();

  // ---- g/theta/phi : [16,256] @ [256,128]^T + bias, wave w -> N-tile w ----
  {
    const float* Ws[3] = {Wg, Wt, Wp};
    const float* bs[3] = {bg, bt, bp};
    float(*outs[3])[INTER] = {g_lds, th_lds, ph_lds};
    const int n = wave * 16 + l;
#pragma unroll
    for (int m = 0; m < 3; ++m) {
      const float* Wm = Ws[m];
      v8f c8 = {};
      for (int k0 = 0; k0 < CC; k0 += 4) {
        v2f a, bfr;
        a.x   = p_lds[l][k0 + 2 * half];
        a.y   = p_lds[l][k0 + 2 * half + 1];
        bfr.x = Wm[n * CC + k0 + 2 * half];      // B[k][n] = W[n][k]
        bfr.y = Wm[n * CC + k0 + 2 * half + 1];
        c8 = wmma_f32_16x16x4(a, bfr, c8);
      }
      const float bias = bs[m][n];
#pragma unroll
      for (int v = 0; v < 8; ++v) outs[m][v + 8 * half][n] = c8[v] + bias;
    }
  }
  __syncthreads();

  // ---- f[b] = relu(th.Wf[:128] + ph.Wf[128:] + bf) ----
  if (tid < BB) {
    float acc = bf[0];
    for (int i = 0; i < INTER; ++i)
      acc += th_lds[tid][i] * Wf[i] + ph_lds[tid][i] * Wf[INTER + i];
    f_lds[tid] = fmaxf(acc, 0.f);
  }
  __syncthreads();

  // ---- wy = (f*g) @ Wz^T + bz : M=16,N=256,K=128; wave -> tiles {w, w+8} --
#pragma unroll
  for (int rep = 0; rep < 2; ++rep) {
    const int t = wave + 8 * rep;
    const int n = t * 16 + l;
    v8f c8 = {};
    for (int k0 = 0; k0 < INTER; k0 += 4) {
      v2f a, bfr;
      const float fb = f_lds[l];
      a.x   = fb * g_lds[l][k0 + 2 * half];
      a.y   = fb * g_lds[l][k0 + 2 * half + 1];
      bfr.x = Wz[n * INTER + k0 + 2 * half];
      bfr.y = Wz[n * INTER + k0 + 2 * half + 1];
      c8 = wmma_f32_16x16x4(a, bfr, c8);
    }
    const float bias = bz[n];
#pragma unroll
    for (int v = 0; v < 8; ++v) wy_lds[v + 8 * half][n] = c8[v] + bias;
  }
  __syncthreads();

  // ---- batchnorm over batch + residual + sigmoid (thread = channel) ----
  {
    const int c = tid;
    float mu = 0.f;
    for (int b = 0; b < BB; ++b) mu += wy_lds[b][c];
    mu *= (1.0f / BB);
    float var = 0.f;
    for (int b = 0; b < BB; ++b) {
      const float d = wy_lds[b][c] - mu;
      var += d * d;
    }
    var *= (1.0f / BB);
    const float inv = rsqrtf(var + 1e-5f);
    const float ga = gamma[c], be = beta[c];
    for (int b = 0; b < BB; ++b) {
      const float z = ga * (wy_lds[b][c] - mu) * inv + be + p_lds[b][c];
      wy_lds[b][c] = 1.f / (1.f + expf(-z));   // scores, in place
    }
  }
  __syncthreads();

  // ---- top-3 channels per batch row (ties -> lowest index, like top_k) ----
  if (tid < BB) {
    int i0 = -1, i1 = -1, i2 = -1;
    for (int j = 0; j < 3; ++j) {
      float best = -1.f;
      int besti = 0;
      for (int c = 0; c < CC; ++c) {
        if (c == i0 || c == i1 || c == i2) continue;
        const float v = wy_lds[tid][c];
        if (v > best) { best = v; besti = c; }
      }
      if (j == 0) i0 = besti;
      else if (j == 1) i1 = besti;
      else i2 = besti;
      topk[tid * 3 + j] = besti;
    }
  }
}

// ---------------------------------------------------------------------------
// Kernel 3: gather the 3 selected channel planes -> out channels 1..3
// ---------------------------------------------------------------------------
__global__ __launch_bounds__(256) void nlb_gather(
    const float* __restrict__ x, const int* __restrict__ topk,
    float* __restrict__ out) {
  const int j = blockIdx.x;  // 0..2
  const int b = blockIdx.y;
  const int c = topk[b * 3 + j];
  const float* src = x + ((size_t)b * CC + c) * HWSZ;
  float* dst = out + (size_t)b * 4 * HWSZ + (size_t)(1 + j) * HWSZ;
  for (int i = threadIdx.x * 4; i < HWSZ; i += 256 * 4)
    *(float4*)(dst + i) = *(const float4*)(src + i);
}

extern "C" void kernel_launch(void* const* d_in, const int* in_sizes, int n_in,
                              void* d_out, int out_size, void* d_ws,
                              size_t ws_size, hipStream_t stream) {
  (void)in_sizes; (void)n_in; (void)out_size; (void)ws_size;
  const float* x     = (const float*)d_in[0];
  const float* Wg    = (const float*)d_in[1];
  const float* bg    = (const float*)d_in[2];
  const float* Wt    = (const float*)d_in[3];
  const float* bt    = (const float*)d_in[4];
  const float* Wp    = (const float*)d_in[5];
  const float* bp    = (const float*)d_in[6];
  const float* Wf    = (const float*)d_in[7];
  const float* bf    = (const float*)d_in[8];
  const float* Wz    = (const float*)d_in[9];
  const float* bz    = (const float*)d_in[10];
  const float* gamma = (const float*)d_in[11];
  const float* beta  = (const float*)d_in[12];
  const float* W3    = (const float*)d_in[13];
  float* out = (float*)d_out;

  float* ws       = (float*)d_ws;
  float* partials = ws;                                  // 16*32*256 floats
  int*   topk     = (int*)(ws + BB * CHUNKS * CC);       // 48 ints

  dim3 g1(CHUNKS, BB);
  nlb_pool_eca<<<g1, 128, 0, stream>>>(x, W3, out, partials);

  nlb_tiny<<<1, 256, 0, stream>>>(partials, Wg, bg, Wt, bt, Wp, bp, Wf, bf,
                                  Wz, bz, gamma, beta, topk);

  dim3 g3(3, BB);
  nlb_gather<<<g3, 256, 0, stream>>>(x, topk, out);
}